// GCN_Attention_88630945120523
// MI455X (gfx1250) — compile-verified
//
#include <hip/hip_runtime.h>
#include <math.h>

#define IN_FEAT 64
#define OUT_FEAT 64

typedef float v2f __attribute__((ext_vector_type(2)));
typedef float v8f __attribute__((ext_vector_type(8)));

// Monotonic order-preserving float<->uint encoding for atomic max.
__device__ __forceinline__ unsigned fenc(float f) {
    unsigned u = __float_as_uint(f);
    return (u & 0x80000000u) ? ~u : (u | 0x80000000u);
}
__device__ __forceinline__ float fdec(unsigned u) {
    return __uint_as_float((u & 0x80000000u) ? (u & 0x7FFFFFFFu) : ~u);
}

// Guaranteed single-instruction f32 atomic add at device scope (RMW at L2).
// Non-returning -> tracked by STOREcnt; S_ENDPGM's implicit wait-idle covers it.
__device__ __forceinline__ void gatomic_fadd(float* p, float v) {
    asm volatile("global_atomic_add_f32 %0, %1, off scope:SCOPE_DEV"
                 :: "v"(p), "v"(v)
                 : "memory");
}

__global__ __launch_bounds__(256) void init_kernel(unsigned* __restrict__ m_enc,
                                                   float* __restrict__ denom, int n) {
    int i = blockIdx.x * blockDim.x + threadIdx.x;
    if (i < n) { m_enc[i] = 0u; denom[i] = 0.0f; }
}

// One wave computes a 16-row tile of BOTH GEMMs using V_WMMA_F32_16X16X4_F32.
// z = feat @ W_out + b_out  -> ws (z)
// r = feat @ res_W + res_b  -> d_out (residual; scatter adds on top later)
__global__ __launch_bounds__(256) void gemm_node_kernel(
    const float* __restrict__ feat,
    const float* __restrict__ Wout, const float* __restrict__ bout,
    const float* __restrict__ Wres, const float* __restrict__ bres,
    float* __restrict__ z, float* __restrict__ out,
    int n_nodes, int n_tiles)
{
    int wave = (int)((blockIdx.x * blockDim.x + threadIdx.x) >> 5);
    int lane = threadIdx.x & 31;
    if (wave >= n_tiles) return;  // wave-uniform exit, EXEC stays all-ones for WMMA

    const int m0    = wave * 16;
    const int lrow  = lane & 15;        // M (for A) / N (for B,C)
    const int khalf = (lane >> 4) * 2;  // lanes 0-15 -> K{0,1}, lanes 16-31 -> K{2,3}
    const int mhi   = (lane >> 4) * 8;  // C/D: lanes 16-31 hold rows M+8
    const bool full = (m0 + 16 <= n_nodes);  // wave-uniform: store path w/o exec churn

    // ---- load 16 A fragments (16x4 f32 each) covering K=0..63 ----
    int arow = m0 + lrow; if (arow > n_nodes - 1) arow = n_nodes - 1;
    const float* aptr = feat + (size_t)arow * IN_FEAT + khalf;
    v2f af[16];
#pragma unroll
    for (int kk = 0; kk < 16; ++kk) {
        af[kk].x = aptr[kk * 4 + 0];
        af[kk].y = aptr[kk * 4 + 1];
    }

#pragma unroll
    for (int mtx = 0; mtx < 2; ++mtx) {
        const float* W    = (mtx == 0) ? Wout : Wres;
        const float* bias = (mtx == 0) ? bout : bres;
        float*       dp   = (mtx == 0) ? z    : out;
#pragma unroll
        for (int nt = 0; nt < 4; ++nt) {
            const int ncol = nt * 16 + lrow;
            v8f acc = {0.f, 0.f, 0.f, 0.f, 0.f, 0.f, 0.f, 0.f};
#pragma unroll
            for (int kk = 0; kk < 16; ++kk) {
                // B frag (4x16): VGPR0 = row K(khalf), VGPR1 = row K(khalf+1)
                v2f bf;
                bf.x = W[(size_t)(kk * 4 + khalf + 0) * OUT_FEAT + ncol];
                bf.y = W[(size_t)(kk * 4 + khalf + 1) * OUT_FEAT + ncol];
                acc = __builtin_amdgcn_wmma_f32_16x16x4_f32(
                          false, af[kk], false, bf, (short)0, acc, false, false);
            }
            const float bv = bias[ncol];
            float* cp = dp + (size_t)(m0 + mhi) * OUT_FEAT + ncol;
            if (full) {
#pragma unroll
                for (int j = 0; j < 8; ++j)
                    cp[(size_t)j * OUT_FEAT] = acc[j] + bv;
            } else {
#pragma unroll
                for (int j = 0; j < 8; ++j) {
                    if (m0 + mhi + j < n_nodes)
                        cp[(size_t)j * OUT_FEAT] = acc[j] + bv;
                }
            }
        }
    }
}

// Per-node attention pre-scores: s_src = z . attn_w[:64], s_dst = z . attn_w[64:]
__global__ __launch_bounds__(256) void s_kernel(const float* __restrict__ z,
                                                const float* __restrict__ attn_w,
                                                float* __restrict__ s_src,
                                                float* __restrict__ s_dst, int n) {
    int i = blockIdx.x * blockDim.x + threadIdx.x;
    if (i >= n) return;
    const float4* zr = (const float4*)(z + (size_t)i * OUT_FEAT);
    const float4* w  = (const float4*)attn_w;
    float a = 0.f, b = 0.f;
#pragma unroll
    for (int k = 0; k < 16; ++k) {
        float4 zv = zr[k];
        float4 wa = w[k];
        float4 wb = w[k + 16];
        a += zv.x * wa.x + zv.y * wa.y + zv.z * wa.z + zv.w * wa.w;
        b += zv.x * wb.x + zv.y * wb.y + zv.z * wb.z + zv.w * wb.w;
    }
    s_src[i] = a;
    s_dst[i] = b;
}

__global__ __launch_bounds__(256) void edge_max_kernel(
    const int* __restrict__ src, const int* __restrict__ dst,
    const float* __restrict__ s_src, const float* __restrict__ s_dst,
    float* __restrict__ ews, unsigned* __restrict__ m_enc, int ne)
{
    int i = blockIdx.x * blockDim.x + threadIdx.x;
    if (i >= ne) return;
    const float x = s_src[src[i]] + s_dst[dst[i]];
    const float e = 0.5f * x * (1.0f + erff(x * 0.70710678118654752440f));  // exact GELU
    ews[i] = e;
    atomicMax(m_enc + dst[i], fenc(e));  // hardware global_atomic_max_u32
}

__global__ __launch_bounds__(256) void edge_exp_kernel(
    const int* __restrict__ dst, float* __restrict__ ews,
    const unsigned* __restrict__ m_enc, float* __restrict__ denom, int ne)
{
    int i = blockIdx.x * blockDim.x + threadIdx.x;
    if (i >= ne) return;
    const int d = dst[i];
    float m = fdec(m_enc[d]);
    const unsigned mu = __float_as_uint(m);
    if ((mu & 0x7F800000u) == 0x7F800000u) m = 0.0f;  // non-finite guard (matches ref)
    const float ex = expf(ews[i] - m);
    ews[i] = ex;
    gatomic_fadd(denom + d, ex);
}

// 16 lanes per edge; each lane handles 4 consecutive output features (float4).
__global__ __launch_bounds__(256) void edge_scatter_kernel(
    const float* __restrict__ z, const int* __restrict__ src,
    const int* __restrict__ dst, const float* __restrict__ ews,
    const float* __restrict__ denom, float* __restrict__ out, int ne)
{
    const int t = blockIdx.x * 256 + threadIdx.x;
    const int e = t >> 4;
    const int l = t & 15;
    if (e >= ne) return;
    const int s = src[e], d = dst[e];
    const float a = ews[e] / fmaxf(denom[d], 1e-20f);
    const float4 zv = ((const float4*)(z + (size_t)s * OUT_FEAT))[l];
    float* op = out + (size_t)d * OUT_FEAT + l * 4;
    gatomic_fadd(op + 0, a * zv.x);
    gatomic_fadd(op + 1, a * zv.y);
    gatomic_fadd(op + 2, a * zv.z);
    gatomic_fadd(op + 3, a * zv.w);
}

extern "C" void kernel_launch(void* const* d_in, const int* in_sizes, int n_in,
                              void* d_out, int out_size, void* d_ws, size_t ws_size,
                              hipStream_t stream)
{
    const float* feature = (const float*)d_in[0];
    const int*   src     = (const int*)d_in[1];
    const int*   dst     = (const int*)d_in[2];
    const float* W_out   = (const float*)d_in[3];
    const float* b_out   = (const float*)d_in[4];
    const float* attn_w  = (const float*)d_in[5];
    const float* res_W   = (const float*)d_in[6];
    const float* res_b   = (const float*)d_in[7];
    float* out = (float*)d_out;

    const int n  = in_sizes[0] / IN_FEAT;   // 100000
    const int ne = in_sizes[1];             // 1600000

    // workspace layout (all f32-sized slots)
    float*    z      = (float*)d_ws;              // n*64
    float*    s_src  = z + (size_t)n * OUT_FEAT;  // n
    float*    s_dst  = s_src + n;                 // n
    unsigned* m_enc  = (unsigned*)(s_dst + n);    // n
    float*    denom  = (float*)(m_enc + n);       // n
    float*    ews    = denom + n;                 // ne

    const int tiles = (n + 15) / 16;

    init_kernel<<<(n + 255) / 256, 256, 0, stream>>>(m_enc, denom, n);
    gemm_node_kernel<<<(tiles + 7) / 8, 256, 0, stream>>>(
        feature, W_out, b_out, res_W, res_b, z, out, n, tiles);
    s_kernel<<<(n + 255) / 256, 256, 0, stream>>>(z, attn_w, s_src, s_dst, n);
    edge_max_kernel<<<(ne + 255) / 256, 256, 0, stream>>>(src, dst, s_src, s_dst, ews, m_enc, ne);
    edge_exp_kernel<<<(ne + 255) / 256, 256, 0, stream>>>(dst, ews, m_enc, denom, ne);
    const long scatter_threads = (long)ne * 16;
    edge_scatter_kernel<<<(unsigned)((scatter_threads + 255) / 256), 256, 0, stream>>>(
        z, src, dst, ews, denom, out, ne);
}